// DeformationMetaMultiMLP_19602230739477
// MI455X (gfx1250) — compile-verified
//
#include <hip/hip_runtime.h>

typedef float v2f __attribute__((ext_vector_type(2)));
typedef float v8f __attribute__((ext_vector_type(8)));

constexpr int kH    = 8;
constexpr int kN    = 2048;
constexpr int kC    = 128;   // COND_DIM
constexpr int kL    = 64;    // LAYER width
constexpr int kIn   = 3;
constexpr int kOut  = 3;
constexpr int kM    = 32;    // points per workgroup (2 WMMA M-tiles)
constexpr int kHPs  = 66;    // hp row stride (65 used, padded)

// One 32-point tile per workgroup. hp: [32][kHPs], res: [32][64] in LDS.
// afrag0/afrag1: cached fp32 A-fragments of two 16x128 cond tiles.
// Each B fragment (16 W-rows x 4 K) feeds TWO independent WMMAs -> 16 FLOP/B.
// Computes res[m,o] += sum_i hp[m,i] * ( cond[m,:]·W[o*GSZ+i,:] + b[o*GSZ+i] )
template <int GSZ>
__device__ __forceinline__ void run_layer(const float* __restrict__ Wt,
                                          const float* __restrict__ bt,
                                          int O,
                                          const v2f* afrag0, const v2f* afrag1,
                                          float* __restrict__ hp,
                                          float* __restrict__ res,
                                          int lane, int wave)
{
    const int col  = lane & 15;   // output column within chunk / B-matrix N index
    const int half = lane >> 4;   // K-half for A/B frags, M-half for C/D
    const int nchunks = (O + 15) >> 4;

    for (int c = wave; c < nchunks; c += 4) {
        const int o0     = c << 4;
        const int ocount = min(16, O - o0);            // wave-uniform
        const int ocol   = min(col, ocount - 1);       // clamp for partial chunk
        const int orow   = o0 + ocol;

        // C init = generated bias b[oi]; same value in all 8 row-VGPRs of a lane
        const float bval = bt[orow];
        v8f acc0, acc1;
#pragma unroll
        for (int v = 0; v < 8; ++v) { acc0[v] = bval; acc1[v] = bval; }

        // B frag: lane holds B[k][n] with n = col, k = 2*half + v  (k-step s adds 4s)
        const float* wrow = Wt + (size_t)orow * kC + 2 * half;
#pragma unroll
        for (int s = 0; s < 32; ++s) {
            v2f bfrag = *(const v2f*)(wrow + 4 * s);
            acc0 = __builtin_amdgcn_wmma_f32_16x16x4_f32(
                false, afrag0[s], false, bfrag, (short)0, acc0, false, false);
            acc1 = __builtin_amdgcn_wmma_f32_16x16x4_f32(
                false, afrag1[s], false, bfrag, (short)0, acc1, false, false);
        }

        // scatter-apply: w[m, oi] contributes hp[m, oi%GSZ] * w to res[m, oi/GSZ]
        if (col < ocount) {
            const int oi = o0 + col;
            const int og = oi / GSZ;
            const int ii = oi - og * GSZ;
#pragma unroll
            for (int v = 0; v < 8; ++v) {
                const int m = v + 8 * half;            // C/D layout: M = v + 8*(lane/16)
                atomicAdd(&res[m        * kL + og], acc0[v] * hp[m        * kHPs + ii]);
                atomicAdd(&res[(m + 16) * kL + og], acc1[v] * hp[(m + 16) * kHPs + ii]);
            }
        }
    }
}

__device__ __forceinline__ void relu_transition(float* __restrict__ hp,
                                                float* __restrict__ res, int tid)
{
    // hp = [relu(res), 1]; res = 0   (each idx touched by exactly one thread)
    for (int idx = tid; idx < kM * kL; idx += 128) {
        float r = res[idx]; res[idx] = 0.0f;
        hp[(idx >> 6) * kHPs + (idx & 63)] = fmaxf(r, 0.0f);
    }
    if (tid < kM) hp[tid * kHPs + 64] = 1.0f;
}

__global__ __launch_bounds__(128)
void meta_mlp_kernel(const float* __restrict__ x,  const float* __restrict__ cond,
                     const float* __restrict__ W0, const float* __restrict__ b0,
                     const float* __restrict__ W1, const float* __restrict__ b1,
                     const float* __restrict__ W2, const float* __restrict__ b2,
                     const float* __restrict__ W3, const float* __restrict__ b3,
                     float* __restrict__ out)
{
    __shared__ float hp [kM * kHPs];
    __shared__ float res[kM * kL];

    const int tiles = kN / kM;                 // 64
    const int h     = blockIdx.x / tiles;
    const int n0    = (blockIdx.x % tiles) * kM;
    const int tid   = threadIdx.x;
    const int lane  = tid & 31;
    const int wave  = tid >> 5;
    const int col   = lane & 15;
    const int half  = lane >> 4;

    // Cache cond tiles (2 x 16 pts x 128) as fp32 WMMA A-fragments:
    // lane holds A[m][k] with m = col, k = 4*s + 2*half + v
    const float* cbase0 = cond + ((size_t)h * kN + n0)      * kC + (size_t)col * kC + 2 * half;
    const float* cbase1 = cond + ((size_t)h * kN + n0 + 16) * kC + (size_t)col * kC + 2 * half;
    v2f afrag0[32], afrag1[32];
#pragma unroll
    for (int s = 0; s < 32; ++s) {
        afrag0[s] = *(const v2f*)(cbase0 + 4 * s);
        afrag1[s] = *(const v2f*)(cbase1 + 4 * s);
    }

    // Init: hp = [x, 1] (GSZ=4 layout), res = 0
    for (int idx = tid; idx < kM * kHPs; idx += 128) hp[idx]  = 0.0f;
    for (int idx = tid; idx < kM * kL;   idx += 128) res[idx] = 0.0f;
    __syncthreads();
    if (tid < kM) {
        const float* xr = x + ((size_t)h * kN + n0 + tid) * kIn;
        hp[tid * kHPs + 0] = xr[0];
        hp[tid * kHPs + 1] = xr[1];
        hp[tid * kHPs + 2] = xr[2];
        hp[tid * kHPs + 3] = 1.0f;
    }
    __syncthreads();

    const size_t hs = (size_t)h;

    // ---- Layer 0: O = 64*4 = 256, GSZ = 4 ----
    run_layer<4>(W0 + hs * 256 * kC, b0 + hs * 256, 256,
                 afrag0, afrag1, hp, res, lane, wave);
    __syncthreads();
    relu_transition(hp, res, tid);
    __syncthreads();

    // ---- Layer 1: O = 64*65 = 4160, GSZ = 65 ----
    run_layer<65>(W1 + hs * 4160 * kC, b1 + hs * 4160, 4160,
                  afrag0, afrag1, hp, res, lane, wave);
    __syncthreads();
    relu_transition(hp, res, tid);
    __syncthreads();

    // ---- Layer 2: O = 4160, GSZ = 65 ----
    run_layer<65>(W2 + hs * 4160 * kC, b2 + hs * 4160, 4160,
                  afrag0, afrag1, hp, res, lane, wave);
    __syncthreads();
    relu_transition(hp, res, tid);
    __syncthreads();

    // ---- Layer 3: O = 3*65 = 195, GSZ = 65 (partial last chunk) ----
    run_layer<65>(W3 + hs * 195 * kC, b3 + hs * 195, 195,
                  afrag0, afrag1, hp, res, lane, wave);
    __syncthreads();

    // Output (scale 1, center 0): out[h, n0+m, o] = res[m][o], o < 3
    if (tid < kM * kOut) {
        const int m = tid / kOut, o = tid % kOut;
        out[((size_t)h * kN + n0 + m) * kOut + o] = res[m * kL + o];
    }
}

extern "C" void kernel_launch(void* const* d_in, const int* in_sizes, int n_in,
                              void* d_out, int out_size, void* d_ws, size_t ws_size,
                              hipStream_t stream) {
    const float* x    = (const float*)d_in[0];
    const float* cond = (const float*)d_in[1];
    const float* W0   = (const float*)d_in[2];
    const float* b0   = (const float*)d_in[3];
    const float* W1   = (const float*)d_in[4];
    const float* b1   = (const float*)d_in[5];
    const float* W2   = (const float*)d_in[6];
    const float* b2   = (const float*)d_in[7];
    const float* W3   = (const float*)d_in[8];
    const float* b3   = (const float*)d_in[9];
    float* out = (float*)d_out;

    dim3 grid(kH * (kN / kM));   // 512 workgroups, one 32-point tile each
    dim3 block(128);             // 4 waves (wave32): split output columns 4-way
    hipLaunchKernelGGL(meta_mlp_kernel, grid, block, 0, stream,
                       x, cond, W0, b0, W1, b1, W2, b2, W3, b3, out);
}